// NeuralCF_48129403519207
// MI455X (gfx1250) — compile-verified
//
#include <hip/hip_runtime.h>
#include <hip/hip_bf16.h>

typedef float v2f __attribute__((ext_vector_type(2)));
typedef float v8f __attribute__((ext_vector_type(8)));

#define DD 128   // embedding dim

__device__ __forceinline__ float wave_reduce_sum(float v) {
    // wave32 butterfly reduction
    for (int m = 16; m >= 1; m >>= 1) v += __shfl_xor(v, m, 32);
    return v;
}

// ---------------------------------------------------------------- zero fill
__global__ void zero_f32(float* __restrict__ p, long n4 /* float4 count */) {
    long i = (long)blockIdx.x * blockDim.x + threadIdx.x;
    long stride = (long)gridDim.x * blockDim.x;
    float4 z = make_float4(0.f, 0.f, 0.f, 0.f);
    for (; i < n4; i += stride) ((float4*)p)[i] = z;
}

// ------------------------------------------------- edge scatter (aggregate)
// aggr[t][dst][:] += w * x[src][:]   (one wave per edge, float4 per lane)
// gather table (25.6MB) and scatter target (51MB) are both L2-resident.
__global__ void edge_scatter(const int* __restrict__ src, const int* __restrict__ dst,
                             const int* __restrict__ et, const float* __restrict__ ew,
                             const float* __restrict__ x, float* __restrict__ aggr,
                             int E, int N) {
    int lane = threadIdx.x & 31;
    long wid = ((long)blockIdx.x * blockDim.x + threadIdx.x) >> 5;
    long nw  = ((long)gridDim.x * blockDim.x) >> 5;
    for (long e = wid; e < E; e += nw) {
        int s = src[e], d = dst[e], t = et[e];
        float w = ew[e];
        float4 v = *(const float4*)(x + (long)s * DD + lane * 4);
        float* o = aggr + ((long)t * N + d) * DD + lane * 4;
        atomicAdd(o + 0, v.x * w);
        atomicAdd(o + 1, v.y * w);
        atomicAdd(o + 2, v.z * w);
        atomicAdd(o + 3, v.w * w);
    }
}

// ---------------------------------------------------------- RGCN fused GEMM
// C[N,128] = A0@W0 + A1@W1 + A2@W2 + bias   (each A* [N,128], W* [128,128])
// block = 256 threads (8 waves); block tile = 80 rows x 128 cols.
// wave w owns cols [16w,16w+16); holds ONE B fragment per k-step, reused
// across 5 row subtiles (5 v8f accumulators) -> 0.4 vmem per wmma.
__global__ __launch_bounds__(256)
void rgcn_gemm(const float* __restrict__ A0, const float* __restrict__ A1,
               const float* __restrict__ A2,
               const float* __restrict__ W0, const float* __restrict__ W1,
               const float* __restrict__ W2,
               const float* __restrict__ bias, float* __restrict__ C,
               int N, int relu) {
    const int KPAD = 388;                 // 384 + 4: stride%64 = 4, conflict-free
    const int ROWS = 80;
    extern __shared__ float lds[];        // ROWS * KPAD floats (~124 KB)
    int m0 = blockIdx.x * ROWS;
    int tid = threadIdx.x;

    // stage the 80x384 A stripe (cols: [A0 | A1 | A2]) into LDS, float4 grain
    for (int idx = tid; idx < ROWS * 96; idx += 256) {
        int row = idx / 96, gc = (idx % 96) * 4;
        int r = m0 + row; if (r >= N) r = N - 1;
        const float* s = (gc < 128) ? (A0 + (long)r * DD + gc)
                       : (gc < 256) ? (A1 + (long)r * DD + (gc - 128))
                                    : (A2 + (long)r * DD + (gc - 256));
        *(float4*)&lds[row * KPAD + gc] = *(const float4*)s;
    }
    __syncthreads();

    int wave = tid >> 5, lane = tid & 31;
    int n0 = wave * 16;
    int half = lane >> 4, l15 = lane & 15;

    v8f c0 = {}, c1 = {}, c2 = {}, c3 = {}, c4 = {};
    for (int k = 0; k < 384; k += 4) {
        int kk = k + half * 2;
        const float* W = (k < 128) ? W0 : (k < 256) ? W1 : W2;
        int kl = kk & 127;
        v2f b;
        b.x = W[(long)kl * DD + n0 + l15];
        b.y = W[(long)(kl + 1) * DD + n0 + l15];
        const float* la = &lds[l15 * KPAD + kk];
        v2f a0 = *(const v2f*)(la + 0 * 16 * KPAD);
        v2f a1 = *(const v2f*)(la + 1 * 16 * KPAD);
        v2f a2 = *(const v2f*)(la + 2 * 16 * KPAD);
        v2f a3 = *(const v2f*)(la + 3 * 16 * KPAD);
        v2f a4 = *(const v2f*)(la + 4 * 16 * KPAD);
        c0 = __builtin_amdgcn_wmma_f32_16x16x4_f32(false, a0, false, b, (short)0, c0, false, false);
        c1 = __builtin_amdgcn_wmma_f32_16x16x4_f32(false, a1, false, b, (short)0, c1, false, false);
        c2 = __builtin_amdgcn_wmma_f32_16x16x4_f32(false, a2, false, b, (short)0, c2, false, false);
        c3 = __builtin_amdgcn_wmma_f32_16x16x4_f32(false, a3, false, b, (short)0, c3, false, false);
        c4 = __builtin_amdgcn_wmma_f32_16x16x4_f32(false, a4, false, b, (short)0, c4, false, false);
    }

    float bv = bias[n0 + l15];
    v8f acc[5] = {c0, c1, c2, c3, c4};
    for (int j = 0; j < 5; ++j) {
        for (int i = 0; i < 8; ++i) {
            int row = m0 + 16 * j + i + half * 8;
            if (row < N) {
                float v = acc[j][i] + bv;
                if (relu) v = fmaxf(v, 0.f);
                C[(long)row * DD + n0 + l15] = v;
            }
        }
    }
}

// ---------------------------------------------------------------- MLP GEMM
// C[M,NOUT] = relu?(A[M,K] @ W[K,NOUT] + bias)
// block = 2*NOUT threads (NOUT/16 waves); block tile = 64 rows x NOUT cols;
// each wave reuses one B fragment across 4 row subtiles.
template <int K, int NOUT>
__global__ void mlp_gemm(const float* __restrict__ A, const float* __restrict__ W,
                         const float* __restrict__ bias, float* __restrict__ C,
                         int relu) {
    const int KPAD = K + 4;               // stride%64 = 4, conflict-free
    const int ROWS = 64;
    extern __shared__ float lds[];        // ROWS * KPAD floats
    int m0 = blockIdx.x * ROWS;
    int tid = threadIdx.x;
    const int T = 2 * NOUT;

    for (int idx = tid; idx < ROWS * (K / 4); idx += T) {
        int row = idx / (K / 4);
        int gc = (idx % (K / 4)) * 4;
        *(float4*)&lds[row * KPAD + gc] = *(const float4*)(A + (long)(m0 + row) * K + gc);
    }
    __syncthreads();

    int wave = tid >> 5, lane = tid & 31;
    int n0 = wave * 16;
    int half = lane >> 4, l15 = lane & 15;

    v8f c0 = {}, c1 = {}, c2 = {}, c3 = {};
    for (int k = 0; k < K; k += 4) {
        int kk = k + half * 2;
        v2f b;
        b.x = W[(long)kk * NOUT + n0 + l15];
        b.y = W[(long)(kk + 1) * NOUT + n0 + l15];
        const float* la = &lds[l15 * KPAD + kk];
        v2f a0 = *(const v2f*)(la + 0 * 16 * KPAD);
        v2f a1 = *(const v2f*)(la + 1 * 16 * KPAD);
        v2f a2 = *(const v2f*)(la + 2 * 16 * KPAD);
        v2f a3 = *(const v2f*)(la + 3 * 16 * KPAD);
        c0 = __builtin_amdgcn_wmma_f32_16x16x4_f32(false, a0, false, b, (short)0, c0, false, false);
        c1 = __builtin_amdgcn_wmma_f32_16x16x4_f32(false, a1, false, b, (short)0, c1, false, false);
        c2 = __builtin_amdgcn_wmma_f32_16x16x4_f32(false, a2, false, b, (short)0, c2, false, false);
        c3 = __builtin_amdgcn_wmma_f32_16x16x4_f32(false, a3, false, b, (short)0, c3, false, false);
    }

    float bv = bias[n0 + l15];
    v8f acc[4] = {c0, c1, c2, c3};
    for (int j = 0; j < 4; ++j) {
        for (int i = 0; i < 8; ++i) {
            int row = m0 + 16 * j + i + half * 8;
            float v = acc[j][i] + bv;
            if (relu) v = fmaxf(v, 0.f);
            C[(long)row * NOUT + n0 + l15] = v;
        }
    }
}

// -------------------------------------------------- ReLU + LayerNorm (rows)
__global__ void relu_ln(float* __restrict__ h, const float* __restrict__ g,
                        const float* __restrict__ be, int N) {
    int lane = threadIdx.x & 31;
    long wid = ((long)blockIdx.x * blockDim.x + threadIdx.x) >> 5;
    long nw  = ((long)gridDim.x * blockDim.x) >> 5;
    for (long row = wid; row < N; row += nw) {
        float4 v = *(const float4*)(h + row * DD + lane * 4);
        v.x = fmaxf(v.x, 0.f); v.y = fmaxf(v.y, 0.f);
        v.z = fmaxf(v.z, 0.f); v.w = fmaxf(v.w, 0.f);
        float mu = wave_reduce_sum(v.x + v.y + v.z + v.w) * (1.f / DD);
        float dx = v.x - mu, dy = v.y - mu, dz = v.z - mu, dw = v.w - mu;
        float var = wave_reduce_sum(dx * dx + dy * dy + dz * dz + dw * dw) * (1.f / DD);
        float r = rsqrtf(var + 1e-5f);
        int b = lane * 4;
        float4 o;
        o.x = dx * r * g[b + 0] + be[b + 0];
        o.y = dy * r * g[b + 1] + be[b + 1];
        o.z = dz * r * g[b + 2] + be[b + 2];
        o.w = dw * r * g[b + 3] + be[b + 3];
        *(float4*)(h + row * DD + lane * 4) = o;
    }
}

// --------------------------------- gather pairs, GMF, and build MLP input z0
__global__ void pair_gather(const int* __restrict__ ui, const int* __restrict__ ii,
                            const float* __restrict__ h2, float* __restrict__ z0,
                            float* __restrict__ gmf, int B) {
    int lane = threadIdx.x & 31;
    long wid = ((long)blockIdx.x * blockDim.x + threadIdx.x) >> 5;
    long nw  = ((long)gridDim.x * blockDim.x) >> 5;
    for (long b = wid; b < B; b += nw) {
        int u = ui[b], it = ii[b];
        float4 a = *(const float4*)(h2 + (long)u  * DD + lane * 4);
        float4 c = *(const float4*)(h2 + (long)it * DD + lane * 4);
        float nu = wave_reduce_sum(a.x * a.x + a.y * a.y + a.z * a.z + a.w * a.w);
        float ni = wave_reduce_sum(c.x * c.x + c.y * c.y + c.z * c.z + c.w * c.w);
        float inv = 1.f / (fmaxf(sqrtf(nu), 1e-12f) * fmaxf(sqrtf(ni), 1e-12f));
        float4 gm;
        gm.x = a.x * c.x * inv; gm.y = a.y * c.y * inv;
        gm.z = a.z * c.z * inv; gm.w = a.w * c.w * inv;
        *(float4*)(gmf + b * DD + lane * 4) = gm;
        *(float4*)(z0 + b * (2 * DD) + lane * 4) = a;
        *(float4*)(z0 + b * (2 * DD) + DD + lane * 4) = c;
    }
}

// --------------------------------------------------------------- final score
__global__ void score_final(const float* __restrict__ gmf, const float* __restrict__ z3,
                            const float* __restrict__ oW, const float* __restrict__ ob,
                            float* __restrict__ out, int B) {
    int lane = threadIdx.x & 31;
    long wid = ((long)blockIdx.x * blockDim.x + threadIdx.x) >> 5;
    long nw  = ((long)gridDim.x * blockDim.x) >> 5;
    for (long b = wid; b < B; b += nw) {
        float4 g = *(const float4*)(gmf + b * DD + lane * 4);
        float4 w = *(const float4*)(oW + lane * 4);
        float s = g.x * w.x + g.y * w.y + g.z * w.z + g.w * w.w;
        if (lane < 8) {
            float4 z  = *(const float4*)(z3 + b * 32 + lane * 4);
            float4 wz = *(const float4*)(oW + DD + lane * 4);
            s += z.x * wz.x + z.y * wz.y + z.z * wz.z + z.w * wz.w;
        }
        s = wave_reduce_sum(s);
        if (lane == 0) out[b] = s + ob[0];
    }
}

extern "C" void kernel_launch(void* const* d_in, const int* in_sizes, int n_in,
                              void* d_out, int out_size, void* d_ws, size_t ws_size,
                              hipStream_t stream) {
    const int*   user  = (const int*)d_in[0];
    const int*   item  = (const int*)d_in[1];
    const int*   eidx  = (const int*)d_in[2];
    const int*   etype = (const int*)d_in[3];
    const float* ew    = (const float*)d_in[4];
    const float* emb   = (const float*)d_in[5];
    const float* W1r   = (const float*)d_in[6];
    const float* W1o   = (const float*)d_in[7];
    const float* b1    = (const float*)d_in[8];
    const float* g1    = (const float*)d_in[9];
    const float* be1   = (const float*)d_in[10];
    const float* W2r   = (const float*)d_in[11];
    const float* W2o   = (const float*)d_in[12];
    const float* b2    = (const float*)d_in[13];
    const float* mW1   = (const float*)d_in[14];
    const float* mb1   = (const float*)d_in[15];
    const float* mW2   = (const float*)d_in[16];
    const float* mb2   = (const float*)d_in[17];
    const float* mW3   = (const float*)d_in[18];
    const float* mb3   = (const float*)d_in[19];
    const float* oW    = (const float*)d_in[20];
    const float* ob    = (const float*)d_in[21];
    float* out = (float*)d_out;

    const int B = in_sizes[0];
    const int E = in_sizes[3];
    const int N = in_sizes[5] / DD;   // 50000

    const int* esrc = eidx;
    const int* edst = eidx + E;

    float* ws = (float*)d_ws;
    size_t ND = (size_t)N * DD;
    float* aggr = ws;              // 2*ND floats
    float* h1   = ws + 2 * ND;     // ND
    float* h2   = ws + 3 * ND;     // ND
    // MLP intermediates alias the aggr region (free after layer-2 GEMM)
    float* z0   = ws;                          // B*256
    float* gmf  = z0 + (size_t)B * 256;        // B*128
    float* z1   = gmf + (size_t)B * DD;        // B*128
    float* z2   = z1 + (size_t)B * DD;         // B*64
    float* z3   = z2 + (size_t)B * 64;         // B*32

    const int rgcnBlocks = (N + 79) / 80;
    const size_t rgcnLds = 80 * 388 * sizeof(float);

    // ---------------- layer 1 ----------------
    zero_f32<<<2048, 256, 0, stream>>>(aggr, (long)(2 * ND) / 4);
    edge_scatter<<<4096, 256, 0, stream>>>(esrc, edst, etype, ew, emb, aggr, E, N);
    rgcn_gemm<<<rgcnBlocks, 256, rgcnLds, stream>>>(aggr, aggr + ND, emb,
                                                    W1r, W1r + DD * DD, W1o, b1, h1, N, 1);
    relu_ln<<<2048, 256, 0, stream>>>(h1, g1, be1, N);   // relu already applied; LN here

    // ---------------- layer 2 ----------------
    zero_f32<<<2048, 256, 0, stream>>>(aggr, (long)(2 * ND) / 4);
    edge_scatter<<<4096, 256, 0, stream>>>(esrc, edst, etype, ew, h1, aggr, E, N);
    rgcn_gemm<<<rgcnBlocks, 256, rgcnLds, stream>>>(aggr, aggr + ND, h1,
                                                    W2r, W2r + DD * DD, W2o, b2, h2, N, 0);

    // ---------------- head ----------------
    pair_gather<<<1024, 256, 0, stream>>>(user, item, h2, z0, gmf, B);
    mlp_gemm<256, 128><<<B / 64, 256, (size_t)64 * 260 * 4, stream>>>(z0, mW1, mb1, z1, 1);
    mlp_gemm<128, 64><<<B / 64, 128, (size_t)64 * 132 * 4, stream>>>(z1, mW2, mb2, z2, 1);
    mlp_gemm<64, 32><<<B / 64, 64, (size_t)64 * 68 * 4, stream>>>(z2, mW3, mb3, z3, 1);
    score_final<<<512, 256, 0, stream>>>(gmf, z3, oW, ob, out, B);
}